// GladLayer_11390253269660
// MI455X (gfx1250) — compile-verified
//
#include <hip/hip_runtime.h>

// ---------------------------------------------------------------------------
// GladLayer for MI455X (gfx1250, wave32).
//
// new_labels[d]  = sum_{e: dst=d} ability[src[e]] @ W_task[etype[e]]
// new_ability[d] = (sum_{e: dst=d} labels[src[e]] @ W_worker[etype[e]]) / deg[d]
//
// Strategy: bucket edges by etype, tile 16 edges at a time, run dense
// 16x64 @ 64x64 f32 GEMMs on the WMMA pipe (V_WMMA_F32_16X16X4_F32, 16
// K-steps), scatter with global f32 atomics, divide by deg at the end.
// ---------------------------------------------------------------------------

typedef __attribute__((ext_vector_type(2))) float v2f;
typedef __attribute__((ext_vector_type(8))) float v8f;

#define FEAT 64
#define MAXR 8
// hdr (ints in d_ws): [0..8) counts, [8..17) offsets (9), [17..25) cursors

__global__ void k_init_hdr(int* hdr) {
    if (threadIdx.x < 64) hdr[threadIdx.x] = 0;
}

__global__ void k_count(const int* __restrict__ etype, int E, int* __restrict__ hdr) {
    int e = blockIdx.x * blockDim.x + threadIdx.x;
    if (e < E) atomicAdd(&hdr[etype[e]], 1);
}

__global__ void k_scan(int* hdr, int R) {
    if (blockIdx.x == 0 && threadIdx.x == 0) {
        int acc = 0;
        for (int r = 0; r < MAXR; ++r) {
            hdr[8 + r] = acc;
            acc += (r < R) ? hdr[r] : 0;
            hdr[17 + r] = 0;            // cursors
        }
        hdr[8 + MAXR] = acc;
    }
}

__global__ void k_bucket(const int* __restrict__ etype, int E,
                         int* __restrict__ hdr, int* __restrict__ sorted) {
    int e = blockIdx.x * blockDim.x + threadIdx.x;
    if (e < E) {
        int r = etype[e];
        int pos = hdr[8 + r] + atomicAdd(&hdr[17 + r], 1);
        sorted[pos] = e;
    }
}

__global__ void k_zero(float* __restrict__ p, int n) {
    for (int i = blockIdx.x * blockDim.x + threadIdx.x; i < n;
         i += gridDim.x * blockDim.x)
        p[i] = 0.0f;
}

// One 16x64 @ 64x64 f32 GEMM on the WMMA pipe + atomic scatter of the
// 16x64 result to out rows selected per edge.
__device__ __forceinline__ void gemm16_scatter(const float* __restrict__ row,
                                               const float* __restrict__ ldsW,
                                               float valid, int dd,
                                               int m, int h,
                                               float* __restrict__ outBase) {
    v8f acc0 = {}, acc1 = {}, acc2 = {}, acc3 = {};
#pragma unroll
    for (int k = 0; k < 16; ++k) {
        const int kk = 4 * k + 2 * h;
        v2f a;
        a.x = row[kk] * valid;          // A: M=m (lane&15), K=kk,kk+1
        a.y = row[kk + 1] * valid;
        const float* w0 = ldsW + kk * FEAT + m;
        const float* w1 = ldsW + (kk + 1) * FEAT + m;
        v2f b0 = {w0[0],  w1[0]};       // B: N=16*nb+m, K=kk,kk+1
        v2f b1 = {w0[16], w1[16]};
        v2f b2 = {w0[32], w1[32]};
        v2f b3 = {w0[48], w1[48]};
        acc0 = __builtin_amdgcn_wmma_f32_16x16x4_f32(false, a, false, b0, (short)0, acc0, false, false);
        acc1 = __builtin_amdgcn_wmma_f32_16x16x4_f32(false, a, false, b1, (short)0, acc1, false, false);
        acc2 = __builtin_amdgcn_wmma_f32_16x16x4_f32(false, a, false, b2, (short)0, acc2, false, false);
        acc3 = __builtin_amdgcn_wmma_f32_16x16x4_f32(false, a, false, b3, (short)0, acc3, false, false);
    }
    // C/D layout: lane n+16h holds rows M = 8h..8h+7 in acc[0..7], col N = n.
#pragma unroll
    for (int v = 0; v < 8; ++v) {
        int rowM  = h * 8 + v;
        int dnode = __shfl(dd, rowM, 32);   // dst of edge rowM (0.0 added if invalid)
        float* o  = outBase + dnode * FEAT + m;
        atomicAdd(o +  0, acc0[v]);
        atomicAdd(o + 16, acc1[v]);
        atomicAdd(o + 32, acc2[v]);
        atomicAdd(o + 48, acc3[v]);
    }
}

__global__ __launch_bounds__(256) void k_gemm_scatter(
    const float* __restrict__ labels, const float* __restrict__ ability,
    const float* __restrict__ Wworker, const float* __restrict__ Wtask,
    const int* __restrict__ src, const int* __restrict__ dst,
    const int* __restrict__ hdr, const int* __restrict__ sorted,
    float* __restrict__ out_labels, float* __restrict__ out_ability,
    int blocksPerRel) {
    __shared__ __align__(16) float ldsWw[FEAT * FEAT];   // 16 KB
    __shared__ __align__(16) float ldsWt[FEAT * FEAT];   // 16 KB

    const int r   = blockIdx.x / blocksPerRel;
    const int blk = blockIdx.x % blocksPerRel;

    // Stage this relation's weights in LDS (float4, 1024 each).
    {
        const float4* sw = (const float4*)(Wworker + r * FEAT * FEAT);
        const float4* st = (const float4*)(Wtask   + r * FEAT * FEAT);
        float4* dw = (float4*)ldsWw;
        float4* dt = (float4*)ldsWt;
        for (int i = threadIdx.x; i < (FEAT * FEAT) / 4; i += blockDim.x) {
            dw[i] = sw[i];
            dt[i] = st[i];
        }
    }
    __syncthreads();

    const int base    = hdr[8 + r];
    const int count   = hdr[r];
    const int ngroups = (count + 15) >> 4;
    const int lane = threadIdx.x & 31;
    const int wave = threadIdx.x >> 5;
    const int m = lane & 15;
    const int h = lane >> 4;
    const int wavesPerRel = blocksPerRel * (blockDim.x >> 5);

    for (int g = blk * (blockDim.x >> 5) + wave; g < ngroups; g += wavesPerRel) {
        const int eslot = g * 16 + m;
        // Branchless tail handling: clamp the index, zero the A row, so
        // EXEC stays all-ones through the WMMAs (ISA requirement).
        const int   sl    = (eslot < count) ? eslot : (count - 1);
        const int   eidx  = sorted[base + sl];
        const float valid = (eslot < count) ? 1.0f : 0.0f;
        const int   s     = src[eidx];
        const int   dd    = dst[eidx];

        const float* labRow = labels  + s * FEAT;
        const float* ablRow = ability + s * FEAT;

        // new_ability_raw[dst] += labels[src] @ W_worker[r]
        gemm16_scatter(labRow, ldsWw, valid, dd, m, h, out_ability);
        // new_labels[dst]      += ability[src] @ W_task[r]
        gemm16_scatter(ablRow, ldsWt, valid, dd, m, h, out_labels);
    }
}

__global__ void k_finalize(float* __restrict__ out_ability,
                           const float* __restrict__ deg, int n) {
    int i = blockIdx.x * blockDim.x + threadIdx.x;
    if (i < n) out_ability[i] = out_ability[i] / deg[i >> 6];
}

extern "C" void kernel_launch(void* const* d_in, const int* in_sizes, int n_in,
                              void* d_out, int out_size, void* d_ws, size_t ws_size,
                              hipStream_t stream) {
    const float* labels  = (const float*)d_in[0];
    const float* ability = (const float*)d_in[1];
    const float* deg     = (const float*)d_in[2];
    const float* Wworker = (const float*)d_in[3];
    const float* Wtask   = (const float*)d_in[4];
    const int*   src     = (const int*)d_in[5];
    const int*   dst     = (const int*)d_in[6];
    const int*   etype   = (const int*)d_in[7];

    const int N = in_sizes[2];                     // deg is [N,1]
    const int E = in_sizes[5];
    int R = in_sizes[3] / (FEAT * FEAT);
    if (R > MAXR) R = MAXR;

    float* out_labels  = (float*)d_out;
    float* out_ability = (float*)d_out + (size_t)N * FEAT;

    int* hdr    = (int*)d_ws;                      // 64 ints
    int* sorted = hdr + 64;                        // E ints

    k_init_hdr<<<1, 64, 0, stream>>>(hdr);
    k_count<<<(E + 255) / 256, 256, 0, stream>>>(etype, E, hdr);
    k_scan<<<1, 1, 0, stream>>>(hdr, R);
    k_bucket<<<(E + 255) / 256, 256, 0, stream>>>(etype, E, hdr, sorted);
    k_zero<<<1024, 256, 0, stream>>>((float*)d_out, 2 * N * FEAT);

    const int blocksPerRel = 64;
    k_gemm_scatter<<<R * blocksPerRel, 256, 0, stream>>>(
        labels, ability, Wworker, Wtask, src, dst, hdr, sorted,
        out_labels, out_ability, blocksPerRel);

    k_finalize<<<(N * FEAT + 255) / 256, 256, 0, stream>>>(out_ability, deg, N * FEAT);
}